// ImagemoveCollisionCost_84842783965226
// MI455X (gfx1250) — compile-verified
//
#include <hip/hip_runtime.h>

// ---------------------------------------------------------------------------
// ImagemoveCollisionCost for MI455X (gfx1250).
// N ~ 1M particles; 12 random f32 gathers/particle into three 16MB maps that
// all fit in the 192MB L2. Streamed input staged via the CDNA5 async-to-LDS
// path (ASYNCcnt); streamed outputs stored non-temporal to protect L2 map
// residency. No matrix math in this op, so no WMMA — async-tensor path is the
// CDNA5-specific feature that applies.
// ---------------------------------------------------------------------------

#define W_IMG 2048
#define H_IMG 2048

#if defined(__AMDGCN__) && __has_builtin(__builtin_amdgcn_global_load_async_to_lds_b128)
#define USE_ASYNC_LDS 1
#else
#define USE_ASYNC_LDS 0
#endif

// Pointee type the builtin expects: 16-byte int vector (per hipcc diagnostic).
typedef int v4i_ __attribute__((vector_size(4 * sizeof(int))));
typedef __attribute__((address_space(1))) v4i_* gptr_b128_t;
typedef __attribute__((address_space(3))) v4i_* lptr_b128_t;

__device__ __forceinline__ float bilin(const float* __restrict__ img, int base,
                                       float fx, float fy) {
  float v00 = img[base];
  float v01 = img[base + 1];
  float v10 = img[base + W_IMG];
  float v11 = img[base + W_IMG + 1];
  float top = v00 * (1.0f - fx) + v01 * fx;
  float bot = v10 * (1.0f - fx) + v11 * fx;
  return top * (1.0f - fy) + bot * fy;
}

__global__ __launch_bounds__(256) void ImagemoveCollisionCost_kernel(
    const float4* __restrict__ state,   // [N] of (x, y, vx, vy)
    const float*  __restrict__ sdf,     // [H_IMG * W_IMG]
    const float*  __restrict__ gym,     // grad_y map
    const float*  __restrict__ gxm,     // grad_x map
    float* __restrict__ out,            // [2*N]: [0,N) cost, [N,2N) judge
    int N) {
  const int tid = threadIdx.x;
  const int gid = blockIdx.x * 256 + tid;
  const int i   = (gid < N) ? gid : (N - 1);   // clamp instead of early-return

  float4 s;
#if USE_ASYNC_LDS
  __shared__ float4 tile[256];
  // CDNA5 async global->LDS copy (GLOBAL_LOAD_ASYNC_TO_LDS_B128, ASYNCcnt).
  __builtin_amdgcn_global_load_async_to_lds_b128(
      (gptr_b128_t)(state + i),
      (lptr_b128_t)(tile + tid),
      /*offset=*/0, /*cpol=*/0);
#if __has_builtin(__builtin_amdgcn_s_wait_asynccnt)
  __builtin_amdgcn_s_wait_asynccnt(0);
#else
  asm volatile("s_wait_asynccnt 0" ::: "memory");
#endif
  // Each lane reads back only its own slot -> no cross-wave barrier needed.
  s = tile[tid];
#else
  s = state[i];
#endif

  const float x  = s.x;
  const float y  = s.y;
  const float vx = s.z;
  const float vy = s.w;

  // world [-1,1] -> pixel coords, clipped (matches reference order of ops)
  float px = fminf(fmaxf((x + 1.0f) * 0.5f * (float)(W_IMG - 1), 0.0f),
                   (float)(W_IMG - 1));
  float py = fminf(fmaxf((y + 1.0f) * 0.5f * (float)(H_IMG - 1), 0.0f),
                   (float)(H_IMG - 1));

  int x0 = (int)floorf(px);
  int y0 = (int)floorf(py);
  x0 = x0 < 0 ? 0 : (x0 > W_IMG - 2 ? W_IMG - 2 : x0);
  y0 = y0 < 0 ? 0 : (y0 > H_IMG - 2 ? H_IMG - 2 : y0);
  const float fx = px - (float)x0;
  const float fy = py - (float)y0;
  const int base = y0 * W_IMG + x0;

  // 12 gathers total, shared addressing across the three maps (L2-resident)
  const float potential = bilin(sdf, base, fx, fy);
  const float grad_y    = bilin(gym, base, fx, fy);
  const float grad_x    = bilin(gxm, base, fx, fy);

  const float vel_abs  = sqrtf(vx * vx + vy * vy);
  const float grad_abs = sqrtf(grad_x * grad_x + grad_y * grad_y);
  const float dot      = vx * grad_x + vy * grad_y;
  const float ct       = dot / (vel_abs * grad_abs + 1e-6f);
  const float neg_cos  = -fminf(fmaxf(ct, -1.0f), 1.0f);

  const float judge = potential + 0.5f * (potential * vel_abs);
  const float factor = 1.0f + fmaxf(neg_cos, 0.0f) + 0.8f * fminf(neg_cos, 0.0f);
  const float cost  = potential + 0.5f * (potential * vel_abs) * factor;

  if (gid < N) {
    // Non-temporal: write-once stream; keep L2 for the gather maps.
    __builtin_nontemporal_store(cost,  &out[gid]);
    __builtin_nontemporal_store(judge, &out[gid + N]);
  }
}

extern "C" void kernel_launch(void* const* d_in, const int* in_sizes, int n_in,
                              void* d_out, int out_size, void* d_ws, size_t ws_size,
                              hipStream_t stream) {
  // setup_inputs order: state_seq [B,T,4], sdf, grad_y_map, grad_x_map
  const float4* state = (const float4*)d_in[0];
  const float*  sdf   = (const float*)d_in[1];
  const float*  gym   = (const float*)d_in[2];
  const float*  gxm   = (const float*)d_in[3];
  float* out = (float*)d_out;

  const int N = in_sizes[0] / 4;              // 8192*128 = 1,048,576
  const int blocks = (N + 255) / 256;         // 4096 -> exact, no divergence

  ImagemoveCollisionCost_kernel<<<blocks, 256, 0, stream>>>(
      state, sdf, gym, gxm, out, N);
}